// ScatteringMapping_18296560681067
// MI455X (gfx1250) — compile-verified
//
#include <hip/hip_runtime.h>

// ScatteringMapping on MI455X (gfx1250, wave32).
// out[t] = R3 @ G_t,  G_t[i,:] = V2buf[t - s2_i][i,:],  V2buf[tau] = R2 @ M_tau,
// M_tau[j,:] = R1[j,k]*U0[k,:] where s0_k = tau - s1_j (s0 values are distinct).
// Memory-bound: ~400MB f32 output; zero-fill with NT b128 stores, compute only
// populated slots (<=32768) with V_WMMA_F32_16X16X4_F32.

#define T2MAX 3072     // provable bound: max(s1)+max(s0) <= 2976+95
#define S0TAB 256      // s0 <= 95
#define G0 0.9999f

typedef __attribute__((ext_vector_type(2))) float v2f;
typedef __attribute__((ext_vector_type(4))) float v4f;
typedef __attribute__((ext_vector_type(8))) float v8f;

// ---- workspace layout (in floats) ----
// [0,1024)=R1  [1024,2048)=R2  [2048,3072)=R3
// ints at ws+3072: s0inv[256], popflag[3072]
// V2buf: ws + 3072 + 256 + 3072, size 3072*1024 floats  (~12.6 MB total)

// 16x16 tile of a 32x32 f32 GEMM: D = A(32x32) * B(32x32), K chained 8x4.
// A frag (ISA 7.12.2): lane(h=lane>>4, m=lane&15): v0=A[m,2h], v1=A[m,2h+1]
// B frag: VGPR p, half h -> row p + 2h (row-striped, half-offset R/2=2)
// C/D: VGPR p, half h -> row p + 8h, col = lane&15
__device__ __forceinline__ v8f wmma_tile32(const float* __restrict__ A,
                                           const float* __restrict__ B,
                                           int mt, int nt, int lane) {
  const int h = lane >> 4;
  const int nm = lane & 15;
  v8f acc = {0.f, 0.f, 0.f, 0.f, 0.f, 0.f, 0.f, 0.f};
#pragma unroll
  for (int kk = 0; kk < 8; ++kk) {
    const int kb = kk * 4 + 2 * h;
    v2f a, b;
    a.x = A[(mt * 16 + nm) * 32 + kb];
    a.y = A[(mt * 16 + nm) * 32 + kb + 1];
    b.x = B[kb * 32 + nt * 16 + nm];
    b.y = B[(kb + 1) * 32 + nt * 16 + nm];
    acc = __builtin_amdgcn_wmma_f32_16x16x4_f32(false, a, false, b,
                                                (short)0, acc, false, false);
  }
  return acc;
}

// ---- zero-fill output: pure HBM stream, nontemporal b128 ----
__global__ void sm_zero(v4f* __restrict__ out, long long n4) {
  long long i = (long long)blockIdx.x * blockDim.x + threadIdx.x;
  const long long stride = (long long)gridDim.x * blockDim.x;
  v4f z = {0.f, 0.f, 0.f, 0.f};
  for (; i < n4; i += stride) __builtin_nontemporal_store(z, out + i);
}

// ---- phase 0: gain-scaled R matrices + s0 inverse table (1 block, 256 thr) ----
__global__ void sm_phase0(const float* __restrict__ U,
                          const int* __restrict__ shifts,
                          float* __restrict__ ws) {
  const int tid = threadIdx.x;
  int* s0inv = (int*)(ws + 3072);
  s0inv[tid] = -1;
  __syncthreads();
  if (tid < 32) {
    const int s0 = shifts[tid];
    if (s0 >= 0 && s0 < S0TAB) s0inv[s0] = tid;  // s0 values are distinct
  }
#pragma unroll
  for (int m = 0; m < 3; ++m) {
    for (int idx = tid; idx < 1024; idx += 256) {
      const int col = idx & 31;
      const float g = powf(G0, (float)shifts[m * 32 + col]);  // column scaling
      ws[m * 1024 + idx] = U[(m + 1) * 1024 + idx] * g;       // R_{m+1}
    }
  }
}

// ---- phase 1: stage-2 tap table. one wave per tau in [0,3072) ----
__global__ void __launch_bounds__(32)
sm_phase1(const float* __restrict__ U, const int* __restrict__ shifts,
          float* __restrict__ ws) {
  __shared__ float Msh[1024];
  __shared__ int ksh[32];
  const int lane = threadIdx.x;
  const int tau = blockIdx.x;
  const int* s0inv = (const int*)(ws + 3072);
  int* popflag = (int*)(ws + 3072 + S0TAB);
  float* V2buf = ws + 3072 + S0TAB + T2MAX;

  const int d = tau - shifts[32 + lane];                 // tau - s1_j, j = lane
  const int kj = (d >= 0 && d < S0TAB) ? s0inv[d] : -1;  // unique k or none
  ksh[lane] = kj;
  const unsigned long long bal = __ballot(kj >= 0);
  if (lane == 0) popflag[tau] = (bal != 0ull) ? 1 : 0;
  __syncthreads();
  if (bal == 0ull) return;  // uniform across the single wave

  const float* R1 = ws;
  const float* R2 = ws + 1024;
  for (int j = 0; j < 32; ++j) {  // M[j,:] = R1[j,k_j]*U0[k_j,:] (or 0)
    const int k = ksh[j];
    Msh[j * 32 + lane] = (k >= 0) ? R1[j * 32 + k] * U[k * 32 + lane] : 0.0f;
  }
  __syncthreads();

  const int h = lane >> 4, nm = lane & 15;
#pragma unroll
  for (int mt = 0; mt < 2; ++mt)
#pragma unroll
    for (int nt = 0; nt < 2; ++nt) {
      v8f acc = wmma_tile32(R2, Msh, mt, nt, lane);  // V2 = R2 @ M
#pragma unroll
      for (int p = 0; p < 8; ++p) {
        const int row = mt * 16 + p + 8 * h;
        const int col = nt * 16 + nm;
        V2buf[(long long)tau * 1024 + row * 32 + col] = acc[p];
      }
    }
}

// ---- phase 2: one wave per (tau, i0) candidate; min-i owner computes out[t] ----
__global__ void __launch_bounds__(256)
sm_phase2(const int* __restrict__ shifts, const int* __restrict__ mL,
          const int* __restrict__ mR, const float* __restrict__ ws,
          float* __restrict__ out) {
  __shared__ float Gsh[8][1024];
  const int lane = threadIdx.x & 31;
  const int wv = threadIdx.x >> 5;
  const int wg = blockIdx.x * 8 + wv;      // [0, 3072*32)
  const int tau = wg >> 5;
  const int i0 = wg & 31;
  const int* popflag = (const int*)(ws + 3072 + S0TAB);
  const float* V2buf = ws + 3072 + S0TAB + T2MAX;
  if (!popflag[tau]) return;

  const int s2l = shifts[64 + lane];
  const int mLl = mL[lane];
  const int mRl = mR[lane];
  const int t = __shfl(s2l, i0, 32) + tau;  // this candidate's output slot

  // owner = smallest i whose tau_i = t - s2_i is a populated stage-2 slot
  for (int i = 0; i < i0; ++i) {
    const int ti = t - __shfl(s2l, i, 32);
    if (ti >= 0 && ti < T2MAX && popflag[ti]) return;  // someone else owns t
  }

  // gather G_t rows from the (L2-resident) stage-2 table
  float* G = Gsh[wv];
  for (int i = 0; i < 32; ++i) {
    const int ti = t - __shfl(s2l, i, 32);
    float v = 0.0f;
    if (ti >= 0 && ti < T2MAX && popflag[ti])
      v = V2buf[(long long)ti * 1024 + i * 32 + lane];
    G[i * 32 + lane] = v;
  }
  asm volatile("s_wait_dscnt 0" ::: "memory");  // LDS writes visible wave-wide

  const float* R3 = ws + 2048;
  const int h = lane >> 4, nm = lane & 15;
#pragma unroll
  for (int mt = 0; mt < 2; ++mt)
#pragma unroll
    for (int nt = 0; nt < 2; ++nt) {
      v8f acc = wmma_tile32(R3, G, mt, nt, lane);  // out[t] = R3 @ G_t
#pragma unroll
      for (int p = 0; p < 8; ++p) {
        const int row = mt * 16 + p + 8 * h;
        const int col = nt * 16 + nm;
        // final per-row/per-col delays (m_L, m_R): injective in t per (row,col)
        const int tEff = t + __shfl(mLl, row, 32) + __shfl(mRl, col, 32);
        out[(long long)tEff * 1024 + row * 32 + col] = acc[p];
      }
    }
}

extern "C" void kernel_launch(void* const* d_in, const int* in_sizes, int n_in,
                              void* d_out, int out_size, void* d_ws,
                              size_t ws_size, hipStream_t stream) {
  const float* U = (const float*)d_in[0];       // [4,32,32] f32
  const int* shifts = (const int*)d_in[1];      // [3,32] i32
  const int* mL = (const int*)d_in[2];          // [32] i32 (zeros)
  const int* mR = (const int*)d_in[3];          // [32] i32 (zeros)
  float* out = (float*)d_out;                   // [T,32,32] f32
  float* ws = (float*)d_ws;                     // needs ~12.61 MB

  const long long n4 = (long long)out_size / 4;
  sm_zero<<<8192, 256, 0, stream>>>((v4f*)d_out, n4);
  sm_phase0<<<1, 256, 0, stream>>>(U, shifts, ws);
  sm_phase1<<<T2MAX, 32, 0, stream>>>(U, shifts, ws);
  sm_phase2<<<(T2MAX * 32) / 8, 256, 0, stream>>>(shifts, mL, mR, ws, out);
}